// MetaPathTransformer_10754598110052
// MI455X (gfx1250) — compile-verified
//
#include <hip/hip_runtime.h>

typedef __attribute__((ext_vector_type(2))) float v2f;
typedef __attribute__((ext_vector_type(8))) float v8f;

#define SZ_B 2
#define SZ_A 9
#define SZ_N 1024
#define SZ_P 4
#define SZ_D 256
#define SZ_H 8
#define SZ_DH 32
#define SZ_NN (SZ_N * SZ_N)

// ---------------- softmax(relu(kernels)) over relation axis ----------------
__global__ void mpt_softmax_kernel(const float* __restrict__ kern, float* __restrict__ soft) {
  int t = threadIdx.x;
  if (t >= SZ_H * SZ_P) return;
  int h = t / SZ_P, p = t % SZ_P;
  float r[SZ_A];
  float m = 0.0f;  // relu output >= 0
  for (int a = 0; a < SZ_A; ++a) {
    float v = kern[(h * SZ_A + a) * SZ_P + p];
    r[a] = v > 0.0f ? v : 0.0f;
    m = fmaxf(m, r[a]);
  }
  float s = 0.0f;
  for (int a = 0; a < SZ_A; ++a) { r[a] = expf(r[a] - m); s += r[a]; }
  float inv = 1.0f / s;
  for (int a = 0; a < SZ_A; ++a) soft[(h * SZ_A + a) * SZ_P + p] = r[a] * inv;
}

// ------- mix[h,b] = sum_a soft[h,a,step] * adjacency[b,a]  (all 8 heads per thread) -------
__global__ __launch_bounds__(256) void mpt_mix_kernel(const float* __restrict__ adj,
                                                      const float* __restrict__ soft,
                                                      float* __restrict__ out, int step) {
  size_t idx = (size_t)blockIdx.x * 256 + threadIdx.x;  // over B*N*N
  int b = (int)(idx >> 20);
  size_t e = idx & (size_t)(SZ_NN - 1);
  float av[SZ_A];
#pragma unroll
  for (int a = 0; a < SZ_A; ++a)
    av[a] = adj[((size_t)(b * SZ_A + a) << 20) + e];
#pragma unroll
  for (int h = 0; h < SZ_H; ++h) {
    float s = 0.0f;
#pragma unroll
    for (int a = 0; a < SZ_A; ++a)
      s += soft[(h * SZ_A + a) * SZ_P + step] * av[a];
    out[((size_t)(h * SZ_B + b) << 20) + e] = s;
  }
}

// ---------------- fp32 WMMA GEMM: Rout = Rin(1024x1024) * Bm(1024x1024), 16 batches ----------------
// Each wave computes a 64x64 C tile: 4x4 accumulators, K in steps of 4 (V_WMMA_F32_16X16X4_F32).
// A frag (16x4):  lane L: row = L%16,  a[v] = A[row][k + 2*(L/16) + v]
// B frag (4x16):  lane L: col = L%16,  b[v] = B[k + 2*(L/16) + v][col]
// C/D (16x16):    lane L: col = L%16,  c[r] = C[r + 8*(L/16)][col]
__global__ __launch_bounds__(256) void mpt_chain_gemm(const float* __restrict__ Rin,
                                                      const float* __restrict__ Bm,
                                                      float* __restrict__ Rout) {
  const int mat = blockIdx.y;
  const float* Ap = Rin + (size_t)mat * SZ_NN;
  const float* Bp = Bm + (size_t)mat * SZ_NN;
  float* Cp = Rout + (size_t)mat * SZ_NN;
  const int wave = threadIdx.x >> 5;
  const int lane = threadIdx.x & 31;
  const int t = blockIdx.x * 8 + wave;   // 0..255 tiles of 64x64
  const int tileM = (t >> 4) << 6;
  const int tileN = (t & 15) << 6;
  const int lm = lane & 15;
  const int half = lane >> 4;
  const int kh = half << 1;

  v8f acc[4][4] = {};
  for (int k = 0; k < SZ_N; k += 4) {
    v2f a[4];
#pragma unroll
    for (int mi = 0; mi < 4; ++mi)
      a[mi] = *(const v2f*)(Ap + (size_t)(tileM + mi * 16 + lm) * SZ_N + k + kh);
    v2f b[4];
#pragma unroll
    for (int ni = 0; ni < 4; ++ni) {
      const float* bp = Bp + (size_t)(k + kh) * SZ_N + tileN + ni * 16 + lm;
      b[ni].x = bp[0];
      b[ni].y = bp[SZ_N];
    }
#pragma unroll
    for (int mi = 0; mi < 4; ++mi)
#pragma unroll
      for (int ni = 0; ni < 4; ++ni)
        acc[mi][ni] = __builtin_amdgcn_wmma_f32_16x16x4_f32(
            false, a[mi], false, b[ni], (short)0, acc[mi][ni], false, false);
  }
#pragma unroll
  for (int mi = 0; mi < 4; ++mi)
#pragma unroll
    for (int ni = 0; ni < 4; ++ni)
#pragma unroll
      for (int r = 0; r < 8; ++r) {
        int m = tileM + mi * 16 + r + 8 * half;
        Cp[(size_t)m * SZ_N + tileN + ni * 16 + lm] = acc[mi][ni][r];
      }
}

// ---------------- heads = diag(1/deg) * R * V : per-batch 1024x1024 @ 1024x32 ----------------
__global__ __launch_bounds__(256) void mpt_heads_gemm(const float* __restrict__ R,
                                                      const float* __restrict__ V,
                                                      const float* __restrict__ degree,
                                                      float* __restrict__ heads) {
  const int mat = blockIdx.y;            // h*B + b
  const int b = mat % SZ_B;
  const float* Ap = R + (size_t)mat * SZ_NN;
  const float* Bp = V + (size_t)mat * SZ_N * SZ_DH;
  float* Cp = heads + (size_t)mat * SZ_N * SZ_DH;
  const int wave = threadIdx.x >> 5;
  const int lane = threadIdx.x & 31;
  const int t = blockIdx.x * 8 + wave;   // 0..15 tiles of 64x32
  const int tileM = t << 6;
  const int lm = lane & 15;
  const int half = lane >> 4;
  const int kh = half << 1;

  v8f acc[4][2] = {};
  for (int k = 0; k < SZ_N; k += 4) {
    v2f a[4];
#pragma unroll
    for (int mi = 0; mi < 4; ++mi)
      a[mi] = *(const v2f*)(Ap + (size_t)(tileM + mi * 16 + lm) * SZ_N + k + kh);
    v2f bv[2];
#pragma unroll
    for (int ni = 0; ni < 2; ++ni) {
      const float* bp = Bp + (size_t)(k + kh) * SZ_DH + ni * 16 + lm;
      bv[ni].x = bp[0];
      bv[ni].y = bp[SZ_DH];
    }
#pragma unroll
    for (int mi = 0; mi < 4; ++mi)
#pragma unroll
      for (int ni = 0; ni < 2; ++ni)
        acc[mi][ni] = __builtin_amdgcn_wmma_f32_16x16x4_f32(
            false, a[mi], false, bv[ni], (short)0, acc[mi][ni], false, false);
  }
#pragma unroll
  for (int mi = 0; mi < 4; ++mi)
#pragma unroll
    for (int r = 0; r < 8; ++r) {
      int m = tileM + mi * 16 + r + 8 * half;
      float sc = 1.0f / degree[((size_t)(b * SZ_N) + m) * SZ_N + m];
#pragma unroll
      for (int ni = 0; ni < 2; ++ni)
        Cp[(size_t)m * SZ_DH + ni * 16 + lm] = acc[mi][ni][r] * sc;
    }
}

// ---------------- V = x @ Wv + Bv (per head) ----------------
__global__ __launch_bounds__(256) void mpt_v_kernel(const float* __restrict__ x,
                                                    const float* __restrict__ Wv,
                                                    const float* __restrict__ Bv,
                                                    float* __restrict__ V) {
  int idx = blockIdx.x * 256 + threadIdx.x;  // H*B*N*DH
  int e = idx & 31;
  int n = (idx >> 5) & (SZ_N - 1);
  int hb = idx >> 15;
  int b = hb % SZ_B;
  int h = hb / SZ_B;
  float acc = Bv[((size_t)(h * SZ_N) + n) * SZ_DH + e];
  const float* xr = x + (size_t)(b * SZ_N + n) * SZ_D;
  const float* wv = Wv + (size_t)h * SZ_D * SZ_DH + e;
  for (int d = 0; d < SZ_D; ++d) acc += xr[d] * wv[d * SZ_DH];
  V[idx] = acc;
}

// ---------------- residual = concat(heads) @ W0 + x ----------------
__global__ __launch_bounds__(256) void mpt_resid_kernel(const float* __restrict__ heads,
                                                        const float* __restrict__ W0,
                                                        const float* __restrict__ x,
                                                        float* __restrict__ out) {
  int idx = blockIdx.x * 256 + threadIdx.x;  // B*N*D
  int d = idx & (SZ_D - 1);
  int row = idx >> 8;        // b*N + n
  int b = row >> 10;
  int n = row & (SZ_N - 1);
  float acc = 0.0f;
  for (int k = 0; k < SZ_D; ++k) {
    int h = k >> 5, e = k & 31;
    acc += heads[((size_t)(h * SZ_B + b) * SZ_N + n) * SZ_DH + e] * W0[k * SZ_D + d];
  }
  out[idx] = acc + x[idx];
}

// ---------------- layernorm (population var + eps inside sqrt, matches jnp) ----------------
__global__ __launch_bounds__(256) void mpt_ln_kernel(const float* __restrict__ in,
                                                     float* __restrict__ out,
                                                     const float* __restrict__ gamma,
                                                     const float* __restrict__ beta, int L) {
  int row = blockIdx.x;
  const float* rp = in + (size_t)row * L;
  float* op = out + (size_t)row * L;
  __shared__ float ssum[256], ssq[256];
  int tid = threadIdx.x;
  float s = 0.0f, q = 0.0f;
  for (int e = tid; e < L; e += 256) { float v = rp[e]; s += v; q += v * v; }
  ssum[tid] = s; ssq[tid] = q;
  __syncthreads();
  for (int off = 128; off > 0; off >>= 1) {
    if (tid < off) { ssum[tid] += ssum[tid + off]; ssq[tid] += ssq[tid + off]; }
    __syncthreads();
  }
  float mean = ssum[0] / (float)L;
  float var = ssq[0] / (float)L - mean * mean;
  float inv = 1.0f / sqrtf(var + 1e-12f);
  for (int e = tid; e < L; e += 256)
    op[e] = (rp[e] - mean) * inv * gamma[e] + beta[e];
}

// ---------------- f = gelu_exact(hn @ W1 + b1) ----------------
__global__ __launch_bounds__(256) void mpt_ffn1_kernel(const float* __restrict__ hn,
                                                       const float* __restrict__ W1,
                                                       const float* __restrict__ b1,
                                                       float* __restrict__ f) {
  int idx = blockIdx.x * 256 + threadIdx.x;  // B*N*2D
  int j = idx & (2 * SZ_D - 1);
  int row = idx >> 9;
  float acc = b1[j];
  const float* hr = hn + (size_t)row * SZ_D;
  for (int d = 0; d < SZ_D; ++d) acc += hr[d] * W1[d * 2 * SZ_D + j];
  f[idx] = 0.5f * acc * (1.0f + erff(acc * 0.7071067811865476f));
}

// ---------------- out = fn @ W2 + b2f + residual ----------------
__global__ __launch_bounds__(256) void mpt_out_kernel(const float* __restrict__ f,
                                                      const float* __restrict__ W2,
                                                      const float* __restrict__ b2f,
                                                      const float* __restrict__ resid,
                                                      float* __restrict__ out) {
  int idx = blockIdx.x * 256 + threadIdx.x;  // B*N*D
  int d = idx & (SZ_D - 1);
  int row = idx >> 8;
  float acc = b2f[d];
  const float* fr = f + (size_t)row * 2 * SZ_D;
  for (int j = 0; j < 2 * SZ_D; ++j) acc += fr[j] * W2[j * SZ_D + d];
  out[idx] = acc + resid[idx];
}

extern "C" void kernel_launch(void* const* d_in, const int* in_sizes, int n_in,
                              void* d_out, int out_size, void* d_ws, size_t ws_size,
                              hipStream_t stream) {
  (void)in_sizes; (void)n_in; (void)out_size; (void)ws_size;
  const float* adjacency = (const float*)d_in[0];
  const float* degree    = (const float*)d_in[1];
  const float* x         = (const float*)d_in[2];
  const float* kernels   = (const float*)d_in[3];
  const float* Wv        = (const float*)d_in[4];
  const float* Bv        = (const float*)d_in[5];
  const float* W0        = (const float*)d_in[6];
  const float* gamma2    = (const float*)d_in[7];
  const float* beta2     = (const float*)d_in[8];
  const float* W1        = (const float*)d_in[9];
  const float* b1        = (const float*)d_in[10];
  const float* gf        = (const float*)d_in[11];
  const float* bf        = (const float*)d_in[12];
  const float* W2        = (const float*)d_in[13];
  const float* b2f       = (const float*)d_in[14];

  float* ws     = (float*)d_ws;
  float* soft   = ws;                                    // 288 (padded to 1024)
  float* Ra     = ws + 1024;                             // 16M floats (64 MB)
  float* Rb     = Ra + (size_t)16 * SZ_NN;               // 16M floats
  float* Mix    = Rb + (size_t)16 * SZ_NN;               // 16M floats
  float* Vbuf   = Mix + (size_t)16 * SZ_NN;              // 512K floats
  float* headsb = Vbuf + (size_t)SZ_H * SZ_B * SZ_N * SZ_DH;
  float* resid  = headsb + (size_t)SZ_H * SZ_B * SZ_N * SZ_DH;
  float* hn     = resid + (size_t)SZ_B * SZ_N * SZ_D;
  float* fbuf   = hn + (size_t)SZ_B * SZ_N * SZ_D;       // 1M floats

  mpt_softmax_kernel<<<1, 32, 0, stream>>>(kernels, soft);

  // R0 = mix(step 0) written directly into Ra
  mpt_mix_kernel<<<(SZ_B * SZ_NN) / 256, 256, 0, stream>>>(adjacency, soft, Ra, 0);

  float* cur = Ra;
  float* nxt = Rb;
  for (int i = 1; i < SZ_P; ++i) {
    mpt_mix_kernel<<<(SZ_B * SZ_NN) / 256, 256, 0, stream>>>(adjacency, soft, Mix, i);
    mpt_chain_gemm<<<dim3(32, 16), 256, 0, stream>>>(cur, Mix, nxt);
    float* tmp = cur; cur = nxt; nxt = tmp;
  }
  // after 3 steps, final chain result is in `cur` (= Rb)

  mpt_v_kernel<<<(SZ_H * SZ_B * SZ_N * SZ_DH) / 256, 256, 0, stream>>>(x, Wv, Bv, Vbuf);
  mpt_heads_gemm<<<dim3(2, 16), 256, 0, stream>>>(cur, Vbuf, degree, headsb);
  mpt_resid_kernel<<<(SZ_B * SZ_N * SZ_D) / 256, 256, 0, stream>>>(headsb, W0, x, resid);
  mpt_ln_kernel<<<SZ_B * SZ_N, 256, 0, stream>>>(resid, hn, gamma2, beta2, SZ_D);
  mpt_ffn1_kernel<<<(SZ_B * SZ_N * 2 * SZ_D) / 256, 256, 0, stream>>>(hn, W1, b1, fbuf);
  mpt_ln_kernel<<<SZ_B * SZ_N, 256, 0, stream>>>(fbuf, fbuf, gf, bf, 2 * SZ_D);
  mpt_out_kernel<<<(SZ_B * SZ_N * SZ_D) / 256, 256, 0, stream>>>(fbuf, W2, b2f, resid, (float*)d_out);
}